// ConvSTFT_35536559407219
// MI455X (gfx1250) — compile-verified
//
#include <hip/hip_runtime.h>
#include <hip/hip_bf16.h>
#include <math.h>

typedef float v2f __attribute__((ext_vector_type(2)));
typedef float v8f __attribute__((ext_vector_type(8)));

#define WIN_LEN   400
#define WIN_INC   100
#define PAD       300            // WIN_LEN - WIN_INC
#define N_FRAMES  1603
#define N_CH      257            // real channels; imag channel = c + 257
#define N_IN      160000
#define N_BATCH   16
#define FRAMES_PER_BLOCK 64      // 4 waves x 16 frames
#define BLOCK_THREADS    128
#define XS_LEN    (WIN_INC * (FRAMES_PER_BLOCK - 1) + WIN_LEN)  // 6700
#define XS_PAD    6704
#define A_ELEMS   (WIN_LEN * 16) // 6400 floats per panel, layout [k*16 + m]

__global__ __launch_bounds__(BLOCK_THREADS)
void convstft_wmma_f32(const float* __restrict__ in,   // (16, 160000)
                       const float* __restrict__ w,    // (514, 1, 400)
                       float* __restrict__ mags,       // (16, 257, 1603)
                       float* __restrict__ phase)      // (16, 257, 1603)
{
    extern __shared__ float smem[];
    float* Ar = smem;                 // real-channel weight panel [k*16 + m]
    float* Ai = smem + A_ELEMS;       // imag-channel weight panel
    float* Xs = smem + 2 * A_ELEMS;   // staged padded input window

    const int c0  = blockIdx.x * 16;               // real-channel tile base (0..256)
    const int t0  = blockIdx.y * FRAMES_PER_BLOCK; // frame tile base
    const int b   = blockIdx.z;                    // batch
    const int tid = threadIdx.x;

    // ---- stage weight panels (real + imag) into LDS, [k*16 + m] layout ----
    for (int idx = tid; idx < 16 * WIN_LEN; idx += BLOCK_THREADS) {
        const int m = idx / WIN_LEN;
        const int k = idx - m * WIN_LEN;
        const int c = c0 + m;
        float vr = 0.0f, vi = 0.0f;
        if (c < N_CH) {
            vr = w[(size_t)c * WIN_LEN + k];
            vi = w[(size_t)(c + N_CH) * WIN_LEN + k];
        }
        Ar[k * 16 + m] = vr;
        Ai[k * 16 + m] = vi;
    }

    // ---- stage input window with implicit zero padding (pad = 300 each side) ----
    const long  sbase = (long)t0 * WIN_INC - PAD;   // input-coord of Xs[0]
    const float* inb  = in + (size_t)b * N_IN;
    for (int s = tid; s < XS_LEN; s += BLOCK_THREADS) {
        const long g = sbase + s;
        Xs[s] = (g >= 0 && g < (long)N_IN) ? inb[g] : 0.0f;
    }
    __syncthreads();

    // ---- per-wave WMMA fragment coordinates (ISA 7.12.2 layouts) ----
    const int lane = tid & 31;
    const int wave = tid >> 5;
    const int half = lane >> 4;     // selects K pair {2h, 2h+1} for A/B frags
    const int mr   = lane & 15;     // A: row m; B: column n
    const int soff = WIN_INC * (wave * 16 + mr);  // base sample of this B column

    v8f accr = {};   // real-channel 16x16 f32 accumulator
    v8f acci = {};   // imag-channel 16x16 f32 accumulator

    // K = 400 = 100 steps of V_WMMA_F32_16X16X4_F32 (x2 for real+imag)
    for (int kk = 0; kk < WIN_LEN / 4; ++kk) {
        const int kb = 4 * kk + 2 * half;
        v2f a_r, a_i, bx;
        a_r.x = Ar[(kb + 0) * 16 + mr];
        a_r.y = Ar[(kb + 1) * 16 + mr];
        a_i.x = Ai[(kb + 0) * 16 + mr];
        a_i.y = Ai[(kb + 1) * 16 + mr];
        bx.x  = Xs[soff + kb + 0];
        bx.y  = Xs[soff + kb + 1];
        // 8 args: (neg_a, A, neg_b, B, c_mod, C, reuse_a, reuse_b)
        accr = __builtin_amdgcn_wmma_f32_16x16x4_f32(
                   false, a_r, false, bx, (short)0, accr, false, false);
        acci = __builtin_amdgcn_wmma_f32_16x16x4_f32(
                   false, a_i, false, bx, (short)0, acci, false, false);
    }

    // ---- fused epilogue: mags = sqrt(max(r^2+i^2, eps)), phase = atan2(i+eps, r+eps)
    const float EPS = 1.19209290e-07f;  // FLT_EPSILON
    const int t = t0 + wave * 16 + mr;  // C/D: N = lane & 15
    if (t < N_FRAMES) {
        #pragma unroll
        for (int r = 0; r < 8; ++r) {
            const int c = c0 + r + 8 * half;  // C/D: M = r + 8*(lane>>4)
            if (c < N_CH) {
                const float rr = accr[r];
                const float ii = acci[r];
                float m2 = fmaxf(rr * rr + ii * ii, EPS);
                const size_t o = ((size_t)b * N_CH + c) * N_FRAMES + t;
                mags[o]  = sqrtf(m2);
                phase[o] = atan2f(ii + EPS, rr + EPS);
            }
        }
    }
}

extern "C" void kernel_launch(void* const* d_in, const int* in_sizes, int n_in,
                              void* d_out, int out_size, void* d_ws, size_t ws_size,
                              hipStream_t stream) {
    (void)in_sizes; (void)n_in; (void)d_ws; (void)ws_size; (void)out_size;

    const float* in = (const float*)d_in[0];   // (16, 160000) fp32
    const float* w  = (const float*)d_in[1];   // (514, 1, 400) fp32

    float* mags  = (float*)d_out;                                  // (16,257,1603)
    float* phase = mags + (size_t)N_BATCH * N_CH * N_FRAMES;       // (16,257,1603)

    const dim3 grid((N_CH + 15) / 16,                               // 17 channel tiles
                    (N_FRAMES + FRAMES_PER_BLOCK - 1) / FRAMES_PER_BLOCK, // 26
                    N_BATCH);                                       // 16
    const dim3 block(BLOCK_THREADS);
    const size_t smem_bytes = (size_t)(2 * A_ELEMS + XS_PAD) * sizeof(float); // 78,016 B

    convstft_wmma_f32<<<grid, block, smem_bytes, stream>>>(in, w, mags, phase);
}